// MessagingLayer_14276471292546
// MI455X (gfx1250) — compile-verified
//
#include <hip/hip_runtime.h>
#include <hip/hip_bf16.h>
#include <math.h>

// ---------------------------------------------------------------------------
// Problem constants (from reference)
// ---------------------------------------------------------------------------
#define NNODES   100000
#define DTEXT    200
#define DDIM     202                 // D
#define ETYPES   6
#define MEDGES   200000
#define NPOS     512
#define DCOL     (DDIM * ETYPES)    // 1212
#define COLGRP   19                  // 19 * 64 = 1216 >= 1212 column coverage
#define MTILES   (NNODES / 16)       // 6250 exact

typedef __attribute__((ext_vector_type(2))) float v2f;
typedef __attribute__((ext_vector_type(8))) float v8f;

// ---------------------------------------------------------------------------
// 0) zero-init output accumulator + bincount
// ---------------------------------------------------------------------------
__global__ void zero_kernel(float* __restrict__ out, size_t n_out,
                            float* __restrict__ bc, size_t n_bc) {
    size_t i = (size_t)blockIdx.x * blockDim.x + threadIdx.x;
    if (i < n_out) out[i] = 0.0f;
    if (i < n_bc)  bc[i]  = 0.0f;
}

// ---------------------------------------------------------------------------
// 1) pos_gating[p, d] = 2*sigmoid( posemb(p,:) @ W_pos[:, d] + b_pos[d] )
//    posemb(p, k) = sin(p * invf[k])        k in [0,100)
//                 = cos(p * invf[k-100])    k in [100,200)
//                 = 0                       k in [200,202)   (skipped)
//    invf[i] = 10000^(-i/100)
// ---------------------------------------------------------------------------
__global__ void pos_gate_kernel(const float* __restrict__ W_pos,
                                const float* __restrict__ b_pos,
                                float* __restrict__ gate) {
    __shared__ float emb[DTEXT];
    int p   = blockIdx.x;
    int tid = threadIdx.x;
    if (tid < 100) {
        float invf = expf(-((float)tid / 100.0f) * 9.210340371976184f); // ln(1e4)
        float s = (float)p * invf;
        emb[tid]       = sinf(s);
        emb[tid + 100] = cosf(s);
    }
    __syncthreads();
    if (tid < DDIM) {
        float acc = b_pos[tid];
        #pragma unroll 4
        for (int k = 0; k < DTEXT; ++k)
            acc = fmaf(emb[k], W_pos[k * DDIM + tid], acc);
        gate[p * DDIM + tid] = 2.0f / (1.0f + expf(-acc));
    }
}

// ---------------------------------------------------------------------------
// 2) prop[N, 1212] = node_states[N, 202] @ W_transform[202, 1212] + b
//    fp32 WMMA: V_WMMA_F32_16X16X4_F32, one wave = 16 rows x 64 cols.
//
//    Fragment layouts (ISA 7.12.2, fp32, wave32):
//      A 16x4 : v2f, lanes 0-15 row M=lane  {K=k0, k0+1};
//                    lanes16-31 row M=lane-16 {K=k0+2, k0+3}
//      B 4x16 : v2f, lane holds column N=lane&15;
//                    lanes 0-15 rows {K=k0,k0+1}, lanes 16-31 {K=k0+2,k0+3}
//      C 16x16: v8f, VGPR r -> row r (lanes 0-15) / row r+8 (lanes 16-31)
// ---------------------------------------------------------------------------
__global__ void gemm_wmma_kernel(const float* __restrict__ A,
                                 const float* __restrict__ W,
                                 const float* __restrict__ bias,
                                 float* __restrict__ C) {
    int wave = (blockIdx.x * blockDim.x + threadIdx.x) >> 5; // wave-uniform
    if (wave >= MTILES * COLGRP) return;                     // uniform exit: EXEC stays full
    int tile_m  = wave / COLGRP;
    int cg      = wave % COLGRP;
    int lane    = threadIdx.x & 31;
    int half    = lane >> 4;      // 0: lanes 0-15, 1: lanes 16-31
    int l16     = lane & 15;
    int colbase = cg * 64;

    // clamp columns: out-of-range lanes accumulate garbage that is never stored
    int c0 = colbase +  0 + l16; c0 = c0 < DCOL ? c0 : DCOL - 1;
    int c1 = colbase + 16 + l16; c1 = c1 < DCOL ? c1 : DCOL - 1;
    int c2 = colbase + 32 + l16; c2 = c2 < DCOL ? c2 : DCOL - 1;
    int c3 = colbase + 48 + l16; c3 = c3 < DCOL ? c3 : DCOL - 1;

    const float* Arow = A + (size_t)(tile_m * 16 + l16) * DDIM;

    v8f acc0 = {}, acc1 = {}, acc2 = {}, acc3 = {};

    // main K loop: k0 = 0..196, all four K's in range
    for (int k0 = 0; k0 < 200; k0 += 4) {
        int ka = k0 + 2 * half;                  // this half-wave's K pair
        v2f a = *(const v2f*)(Arow + ka);        // 8B aligned (202*row + even k)
        const float* Wk = W + (size_t)ka * DCOL;
        v2f b0, b1, b2, b3;
        b0.x = Wk[c0]; b0.y = Wk[DCOL + c0];
        b1.x = Wk[c1]; b1.y = Wk[DCOL + c1];
        b2.x = Wk[c2]; b2.y = Wk[DCOL + c2];
        b3.x = Wk[c3]; b3.y = Wk[DCOL + c3];
        acc0 = __builtin_amdgcn_wmma_f32_16x16x4_f32(false, a, false, b0, (short)0, acc0, false, false);
        acc1 = __builtin_amdgcn_wmma_f32_16x16x4_f32(false, a, false, b1, (short)0, acc1, false, false);
        acc2 = __builtin_amdgcn_wmma_f32_16x16x4_f32(false, a, false, b2, (short)0, acc2, false, false);
        acc3 = __builtin_amdgcn_wmma_f32_16x16x4_f32(false, a, false, b3, (short)0, acc3, false, false);
    }

    // K tail: K = 200,201 are valid rows for every lane, so load them
    // UNCONDITIONALLY (no exec-mask branching) and then cndmask lanes 16-31
    // (which represent K = 202,203) to zero.
    {
        float zmask = (half == 0) ? 1.0f : 0.0f;   // -> v_cndmask, not saveexec
        v2f a, b0, b1, b2, b3;
        float ax  = Arow[200];
        float ay  = Arow[201];
        const float* W200 = W + 200 * DCOL;
        const float* W201 = W + 201 * DCOL;
        float b0x = W200[c0], b0y = W201[c0];
        float b1x = W200[c1], b1y = W201[c1];
        float b2x = W200[c2], b2y = W201[c2];
        float b3x = W200[c3], b3y = W201[c3];
        a.x  = ax  * zmask;  a.y  = ay  * zmask;
        b0.x = b0x * zmask;  b0.y = b0y * zmask;
        b1.x = b1x * zmask;  b1.y = b1y * zmask;
        b2.x = b2x * zmask;  b2.y = b2y * zmask;
        b3.x = b3x * zmask;  b3.y = b3y * zmask;
        acc0 = __builtin_amdgcn_wmma_f32_16x16x4_f32(false, a, false, b0, (short)0, acc0, false, false);
        acc1 = __builtin_amdgcn_wmma_f32_16x16x4_f32(false, a, false, b1, (short)0, acc1, false, false);
        acc2 = __builtin_amdgcn_wmma_f32_16x16x4_f32(false, a, false, b2, (short)0, acc2, false, false);
        acc3 = __builtin_amdgcn_wmma_f32_16x16x4_f32(false, a, false, b3, (short)0, acc3, false, false);
    }

    // store: row = tile_m*16 + r + 8*half, col per lane; guard real columns
    v8f accs[4] = {acc0, acc1, acc2, acc3};
    #pragma unroll
    for (int j = 0; j < 4; ++j) {
        int col = colbase + j * 16 + l16;
        if (col < DCOL) {
            float bv = bias[col];
            #pragma unroll
            for (int r = 0; r < 8; ++r) {
                int orow = tile_m * 16 + r + 8 * half;
                C[(size_t)orow * DCOL + col] = accs[j][r] + bv;
            }
        }
    }
}

// ---------------------------------------------------------------------------
// 3) one wave per edge: gather prop row, gate by position, scatter-add to tgt
// ---------------------------------------------------------------------------
__global__ void scatter_kernel(const float* __restrict__ prop,
                               const float* __restrict__ gate,
                               const int* __restrict__ edges,
                               const int* __restrict__ pos_lists,
                               float* __restrict__ out,
                               float* __restrict__ bincount) {
    int wid = (blockIdx.x * blockDim.x + threadIdx.x) >> 5;
    if (wid >= ETYPES * MEDGES) return;
    int lane = threadIdx.x & 31;
    int e = wid / MEDGES;
    int m = wid - e * MEDGES;
    const int* ep = edges + ((size_t)e * MEDGES + m) * 2;
    int src = ep[0];
    int tgt = ep[1];
    int pos = pos_lists[(size_t)e * MEDGES + m];
    const float* prow = prop + (size_t)src * DCOL + e * DDIM;
    const float* grow = gate + (size_t)pos * DDIM;
    float*       orow = out  + (size_t)tgt * DDIM;
    for (int d = lane; d < DDIM; d += 32)
        atomicAdd(&orow[d], prow[d] * grow[d]);
    if (lane == 0)
        atomicAdd(&bincount[tgt], 1.0f);
}

// ---------------------------------------------------------------------------
// 4) divide each node row by (bincount==0 ? 1 : bincount) + 1e-8
// ---------------------------------------------------------------------------
__global__ void finalize_kernel(float* __restrict__ out,
                                const float* __restrict__ bincount) {
    size_t i = (size_t)blockIdx.x * blockDim.x + threadIdx.x;
    if (i >= (size_t)NNODES * DDIM) return;
    int node = (int)(i / DDIM);
    float c = bincount[node];
    float div = (c == 0.0f ? 1.0f : c) + 1e-8f;
    out[i] = out[i] / div;
}

// ---------------------------------------------------------------------------
// launcher
// ---------------------------------------------------------------------------
extern "C" void kernel_launch(void* const* d_in, const int* in_sizes, int n_in,
                              void* d_out, int out_size, void* d_ws, size_t ws_size,
                              hipStream_t stream) {
    const float* node_states = (const float*)d_in[0];
    const int*   edges       = (const int*)  d_in[1];
    const int*   pos_lists   = (const int*)  d_in[2];
    const float* W_transform = (const float*)d_in[3];
    const float* b_transform = (const float*)d_in[4];
    const float* W_pos       = (const float*)d_in[5];
    const float* b_pos       = (const float*)d_in[6];
    float* out = (float*)d_out;

    char* ws = (char*)d_ws;
    float* gate     = (float*)(ws);                 // 512*202*4   = 413,696 B
    float* bincount = (float*)(ws + 413696);        // 100000*4    = 400,000 B
    float* prop     = (float*)(ws + (1u << 20));    // 100000*1212*4 = 484.8 MB

    // 0) zero accumulators (harness poisons d_out/d_ws with 0xAA)
    size_t n_out = (size_t)NNODES * DDIM;           // 20,200,000
    int zb = (int)((n_out + 255) / 256);
    zero_kernel<<<zb, 256, 0, stream>>>(out, n_out, bincount, (size_t)NNODES);

    // 1) positional gates
    pos_gate_kernel<<<NPOS, 256, 0, stream>>>(W_pos, b_pos, gate);

    // 2) fp32 WMMA GEMM: 6250 row-tiles x 19 col-groups, 8 waves / block
    {
        int total_waves = MTILES * COLGRP;          // 118,750
        int blocks = (total_waves * 32 + 255) / 256;
        gemm_wmma_kernel<<<blocks, 256, 0, stream>>>(node_states, W_transform,
                                                     b_transform, prop);
    }

    // 3) gather/gate/scatter-add, 1 wave per edge
    {
        int total_waves = ETYPES * MEDGES;          // 1,200,000
        int blocks = (total_waves * 32 + 255) / 256;
        scatter_kernel<<<blocks, 256, 0, stream>>>(prop, gate, edges, pos_lists,
                                                   out, bincount);
    }

    // 4) normalize
    finalize_kernel<<<zb, 256, 0, stream>>>(out, bincount);
}